// StarRecognizer_14705968021760
// MI455X (gfx1250) — compile-verified
//
#include <hip/hip_runtime.h>

// ---------------------------------------------------------------------------
// Star-CTC loss for MI455X (gfx1250, wave32).
//   Kernel 0: zero-pad W (V=56 rows) into Wpad (64 rows) -> branchless GEMM
//   Kernel 1: fused logits GEMM (v_wmma_f32_16x16x4_f32) + log_softmax
//   Kernel 2: Star-CTC forward DP, TDM (tensor_load_to_lds) row staging
//   Kernel 3: mean reduction
// ---------------------------------------------------------------------------

typedef __attribute__((ext_vector_type(2))) float        v2f;
typedef __attribute__((ext_vector_type(4))) float        v4f;
typedef __attribute__((ext_vector_type(8))) float        v8f;
typedef __attribute__((ext_vector_type(4))) unsigned int v4u;
typedef __attribute__((ext_vector_type(4))) int          v4i;
typedef __attribute__((ext_vector_type(8))) int          v8i;

#define B_ 32
#define T_ 2048
#define S_ 256
#define D_ 1024
#define V_ 56
#define L_ (2 * S_ + 1)      // 513 extended states
#define NPAD 64              // padded vocab (4 WMMA N-tiles)
#define ROWP 64              // emission row: 56 logp + [56]=star emit
#define NEGF (-1e30f)
#define STAR_PEN (-1.0f)

// ===========================================================================
// Kernel 0: Wpad(64, D) = [W(56, D); zeros(8, D)]   (one float4 per thread)
// ===========================================================================
__global__ __launch_bounds__(256) void pad_w_kernel(
    const float* __restrict__ W, float* __restrict__ Wpad)
{
    const int i   = blockIdx.x * 256 + threadIdx.x;   // float4 index
    const int row = i >> 8;                           // 256 float4 per row
    v4f v = {0.0f, 0.0f, 0.0f, 0.0f};
    if (row < V_) v = *(const v4f*)(W + (size_t)i * 4);
    *(v4f*)(Wpad + (size_t)i * 4) = v;
}

// ===========================================================================
// Kernel 1: logp = log_softmax(feat @ Wpad^T + b), star emission in slot 56.
// 8 waves/block, one 16-row M-tile per wave, 4 N-tiles of 16.
// A-frag per ISA 16x4 f32 layout: lanes0-15 K={hk,hk+1}, hk=(lane>=16)?2:0.
// ===========================================================================
__global__ __launch_bounds__(256) void fused_logits_logsoftmax_kernel(
    const float* __restrict__ feat,   // (B*T, D)
    const float* __restrict__ Wpad,   // (64, D), rows >= 56 zero
    const float* __restrict__ bias,   // (V)
    float* __restrict__ logp)         // (B*T, 64)
{
    __shared__ float sC[8][16][65];   // +1 pad column -> conflict-free rows
    __shared__ float sbias[64];

    const int tid  = threadIdx.x;
    const int wave = tid >> 5;
    const int lane = tid & 31;
    if (tid < 64) sbias[tid] = (tid < V_) ? bias[tid] : 0.0f;

    const int mtile = blockIdx.x * 8 + wave;       // 4096 M-tiles total
    const int row0  = mtile * 16;
    const int r     = lane & 15;                   // matrix row within tile
    const int hk    = (lane >> 4) * 2;             // K sub-offset per half-wave
    const int colb  = lane & 15;                   // N within an N-tile

    const float* aptr = feat + (size_t)(row0 + r) * D_ + hk;
    const float* bptr = Wpad + (size_t)colb * D_ + hk;   // +nt*16*D per tile

    v8f acc[4] = {v8f{}, v8f{}, v8f{}, v8f{}};

    for (int k = 0; k < D_; k += 4) {
        const v2f a = *(const v2f*)(aptr + k);
#pragma unroll
        for (int nt = 0; nt < 4; ++nt) {
            const v2f bf = *(const v2f*)(bptr + (size_t)nt * 16 * D_ + k);
            acc[nt] = __builtin_amdgcn_wmma_f32_16x16x4_f32(
                /*neg_a=*/false, a, /*neg_b=*/false, bf,
                /*c_mod=*/(short)0, acc[nt],
                /*reuse_a=*/false, /*reuse_b=*/false);
        }
    }

    // C/D layout: VGPR j holds M=j (lanes 0-15) / M=j+8 (lanes 16-31), N=lane&15
#pragma unroll
    for (int nt = 0; nt < 4; ++nt)
#pragma unroll
        for (int j = 0; j < 8; ++j)
            sC[wave][j + (lane >> 4) * 8][nt * 16 + colb] = acc[nt][j];

    __syncthreads();   // sC is wave-local but sbias is block-shared

    if (lane < 16) {
        float* rowv = &sC[wave][lane][0];
        float mx = NEGF;
#pragma unroll 8
        for (int n = 0; n < V_; ++n) {
            const float v = rowv[n] + sbias[n];
            rowv[n] = v;
            mx = fmaxf(mx, v);
        }
        float s = 0.0f;
#pragma unroll 8
        for (int n = 0; n < V_; ++n) s += __expf(rowv[n] - mx);
        const float logZ = mx + __logf(s);
        const float m2   = mx - logZ;          // max of normalized logp

        float* dst = logp + (size_t)(row0 + lane) * ROWP;
        float s2 = 0.0f;
#pragma unroll
        for (int n0 = 0; n0 < V_; n0 += 4) {   // 14 x b128 stores
            v4f o;
#pragma unroll
            for (int j = 0; j < 4; ++j) {
                const float lp = rowv[n0 + j] - logZ;
                o[j] = lp;
                s2 += __expf(lp - m2);
            }
            *(v4f*)(dst + n0) = o;
        }
        const float star = STAR_PEN + m2 + __logf(s2);  // penalty baked in
        const v4f t0 = {star, NEGF, NEGF, NEGF};
        const v4f t1 = {NEGF, NEGF, NEGF, NEGF};
        *(v4f*)(dst + 56) = t0;
        *(v4f*)(dst + 60) = t1;
    }
}

// ===========================================================================
// TDM: async-load one 256B emission row (64 f32) into LDS.
// D# group0: count=1 | lds_addr | global_addr[56:0] | type=2.
// D# group1: data_size=4B, tensor_dim0=64, tensor_dim1=1, tile_dim0=64,
//            tile_dim1=1, tensor_dim0_stride=64, tensor_dim1_stride=64.
// Wave-uniform; EXEC ignored by TDM; tracked with TENSORcnt.
// ===========================================================================
__device__ __forceinline__ void tdm_load_row(unsigned lds_addr,
                                             const float* __restrict__ src)
{
    const unsigned long long ga = (unsigned long long)(const void*)src;
    const v4u g0 = {1u,                                   // count=1, user mode
                    lds_addr,                             // bits 63:32
                    (unsigned)(ga & 0xFFFFFFFFu),         // bits 95:64
                    (unsigned)(((ga >> 32) & 0x01FFFFFFu) // bits 120:96
                               | 0x80000000u)};           // type=2 (bits 127:126)
    const v8i g1 = {0x00020000,   // data_size=2 (4 bytes), no multicast/pad
                    0x00400000,   // tensor_dim0 = 64 (bits 79:48, low half)
                    0x00010000,   // tensor_dim0 hi=0, tensor_dim1 = 1
                    0x00400000,   // tensor_dim1 hi=0, tile_dim0 = 64
                    1,            // tile_dim1 = 1, tile_dim2 = 0
                    64,           // tensor_dim0_stride = 64 (low 32)
                    0x00400000,   // stride0 hi=0, tensor_dim1_stride = 64 (lo)
                    0};           // stride1 hi = 0
    const v4i z4 = {0, 0, 0, 0};
#if defined(__clang_major__) && (__clang_major__ >= 23)
    const v8i z8 = {0, 0, 0, 0, 0, 0, 0, 0};
    __builtin_amdgcn_tensor_load_to_lds(g0, g1, z4, z4, z8, 0);
#else
    __builtin_amdgcn_tensor_load_to_lds(g0, g1, z4, z4, 0);
#endif
}

// ===========================================================================
// Kernel 2: Star-CTC forward DP. One block per batch row, 544 threads
// (17 waves), thread l owns extended state l (l < 513). Alpha double-buffered
// in LDS with 2 leading NEGF pad slots. Emission rows double-buffered in LDS,
// staged by wave 0 via TENSOR_LOAD_TO_LDS one step ahead; one barrier/step.
// ===========================================================================
__global__ __launch_bounds__(544) void star_ctc_dp_kernel(
    const float* __restrict__ logp,      // (B*T, 64)
    const int*   __restrict__ targets,   // (B, S)
    const int*   __restrict__ in_len,    // (B)
    const int*   __restrict__ tgt_len,   // (B)
    float*       __restrict__ partials)  // (B) : loss_b / target_len_b
{
    __shared__ float A[2][L_ + 2];
    __shared__ float row[2][ROWP];

    const int b   = blockIdx.x;
    const int l   = threadIdx.x;
    const int wid = threadIdx.x >> 5;
    const int len = in_len[b];
    const int tl  = tgt_len[b];
    const float* base = logp + (size_t)b * T_ * ROWP;

    const unsigned lds0 = (unsigned)(unsigned long long)(void*)&row[0][0];
    const unsigned lds1 = (unsigned)(unsigned long long)(void*)&row[1][0];

    if (wid == 0) tdm_load_row(lds0, base);   // prefetch row t=0

    int  extv  = V_;       // even states read slot 56 (star)
    bool allow = false;
    if (l < L_ && (l & 1)) {
        const int j = l >> 1;
        extv  = targets[b * S_ + j];
        allow = (j == 0) || (extv != targets[b * S_ + j - 1]);
    }
    if (l < 2) { A[0][l] = NEGF; A[1][l] = NEGF; }   // permanent pad slots

    for (int t = 0; t < T_; ++t) {
        const int cur = t & 1;
        if (wid == 0) __builtin_amdgcn_s_wait_tensorcnt(0);  // row t arrived
        __syncthreads();   // row t + prev alpha visible; row buf cur^1 free
        if (wid == 0 && t + 1 < T_)
            tdm_load_row(cur ? lds0 : lds1, base + (size_t)(t + 1) * ROWP);
        if (l < L_) {
            const float e = row[cur][extv];
            float nv;
            if (t == 0) {
                nv = (l < 2) ? e : NEGF;
            } else {
                const int prv = cur ^ 1;
                const float a0 = A[prv][l + 2];
                const float a1 = A[prv][l + 1];
                const float a2 = allow ? A[prv][l] : NEGF;
                const float m  = fmaxf(a0, fmaxf(a1, a2));
                const float s  = __expf(a0 - m) + __expf(a1 - m) +
                                 __expf(a2 - m);
                nv = m + __logf(s) + e;
                nv = (t < len) ? nv : a0;    // freeze past input_length
            }
            A[cur][l + 2] = nv;
        }
    }
    __syncthreads();

    if (threadIdx.x == 0) {
        const int fb   = (T_ - 1) & 1;
        const int last = 2 * tl;
        const float x = A[fb][last + 2];
        const float y = A[fb][last + 1];
        const float m = fmaxf(x, y);
        const float score = m + __logf(__expf(x - m) + __expf(y - m));
        partials[b] = -score / (float)tl;
    }
}

// ===========================================================================
// Kernel 3: mean over B partial losses (d_out is poisoned -> always write).
// ===========================================================================
__global__ void reduce_mean_kernel(const float* __restrict__ partials,
                                   float* __restrict__ out)
{
    if (threadIdx.x == 0) {
        float s = 0.0f;
        for (int i = 0; i < B_; ++i) s += partials[i];
        out[0] = s / (float)B_;
    }
}

// ===========================================================================
extern "C" void kernel_launch(void* const* d_in, const int* in_sizes, int n_in,
                              void* d_out, int out_size, void* d_ws, size_t ws_size,
                              hipStream_t stream)
{
    const float* features = (const float*)d_in[0];   // (B, T, D) f32
    const float* W        = (const float*)d_in[1];   // (V, D)    f32
    const float* bias     = (const float*)d_in[2];   // (V)       f32
    const int*   targets  = (const int*)d_in[3];     // (B, S)    i32
    const int*   in_len   = (const int*)d_in[4];     // (B)       i32
    const int*   tgt_len  = (const int*)d_in[5];     // (B)       i32
    float*       out      = (float*)d_out;

    float* logp     = (float*)d_ws;                       // B*T*64 f32 = 16 MB
    float* partials = logp + (size_t)B_ * T_ * ROWP;      // +32 f32
    float* Wpad     = partials + 32;                      // 64*1024 f32 (16B-aligned)

    pad_w_kernel<<<(NPAD * D_ / 4) / 256, 256, 0, stream>>>(W, Wpad);

    // 4096 M-tiles / 8 waves per block
    fused_logits_logsoftmax_kernel<<<(B_ * T_ / 16) / 8, 256, 0, stream>>>(
        features, Wpad, bias, logp);

    star_ctc_dp_kernel<<<B_, 544, 0, stream>>>(logp, targets, in_len, tgt_len,
                                               partials);

    reduce_mean_kernel<<<1, 32, 0, stream>>>(partials, out);
}